// ScaledDotProductAttention_64536178590057
// MI455X (gfx1250) — compile-verified
//
#include <hip/hip_runtime.h>
#include <hip/hip_bf16.h>

typedef __attribute__((ext_vector_type(2))) float v2f;
typedef __attribute__((ext_vector_type(4))) float v4f;
typedef __attribute__((ext_vector_type(8))) float v8f;

#define S_LEN  1024
#define D_HEAD 64
#define LDS_STRIDE 1028   // 1028 % 64 == 4 -> conflict-free ds_load_b64 A-frags

// ---------------------------------------------------------------------------
// Fully fused residual attention, one workgroup per (bh, 16-row strip):
//   stage 1: S = QK^T*scale + prev + mask   (scores kept in registers,
//            written to global scores output)
//   stage 2: in-register softmax (shuffle + LDS cross-wave reductions),
//            weights written to global + staged in LDS as A-fragments
//   stage 3: O = W @ V  split-K: wave w and w+4 each do half of K=1024 for
//            output tile (w&3); partials combined through LDS (fixed order).
// Block = 256 threads (8 waves); wave w owns score cols [w*128, w*128+128).
// Per-wave live state: sc[8] (64 VGPRs) + afrag (32) -> stays under 256 VGPRs.
// ---------------------------------------------------------------------------
__global__ __launch_bounds__(256) void fused_attn_kernel(
    const float* __restrict__ q, const float* __restrict__ k,
    const float* __restrict__ v, const float* __restrict__ prev,
    const float* __restrict__ mask, const float* __restrict__ scale_p,
    float* __restrict__ out, float* __restrict__ weights,
    float* __restrict__ scores)
{
    extern __shared__ float smem[];
    float* lds_w  = smem;                      // [16][LDS_STRIDE] weight tile
    float* redmax = smem + 16 * LDS_STRIDE;    // [16][8]
    float* redsum = redmax + 128;              // [16][8]
    float* obuf   = redsum + 128;              // [4][16][16] split-K partials

    const int lane = threadIdx.x & 31;
    const int wave = threadIdx.x >> 5;         // 0..7
    const int bh   = blockIdx.y;               // 0..63 (B*H)
    const int m0   = blockIdx.x * 16;          // global row base
    const int hl   = lane >> 4;                // WMMA half (lanes 0-15 / 16-31)
    const int l16  = lane & 15;
    const float scale = scale_p[0];

    // ---------------- stage 1: scores -----------------------------------
    // A-fragments for full D=64: lane holds q[m0+l16][kk*4 + 2*hl + j]
    const float* qrow = q + ((size_t)bh * S_LEN + (m0 + l16)) * D_HEAD;
    v2f afrag[16];
#pragma unroll
    for (int kk = 0; kk < 16; ++kk)
        afrag[kk] = *(const v2f*)(qrow + kk * 4 + hl * 2);

    const float* kbase = k + (size_t)bh * D_HEAD * S_LEN;   // k is [D, S]

    v8f   sc[8];                       // 8 score tiles live (64 VGPRs)
    float rmax[8];
#pragma unroll
    for (int r = 0; r < 8; ++r) rmax[r] = -3.402823466e38f;

#pragma unroll
    for (int nt = 0; nt < 8; ++nt) {
        const int col = wave * 128 + nt * 16 + l16;
        v8f acc = {};
#pragma unroll
        for (int kk = 0; kk < 16; ++kk) {
            const int d = kk * 4 + hl * 2;
            v2f b;
            b.x = kbase[(size_t)d * S_LEN + col];
            b.y = kbase[(size_t)(d + 1) * S_LEN + col];
            acc = __builtin_amdgcn_wmma_f32_16x16x4_f32(
                false, afrag[kk], false, b, (short)0, acc, false, false);
        }
        // epilogue: scale + prev + mask, emit scores, track row max
#pragma unroll
        for (int r = 0; r < 8; ++r) {
            const int row = m0 + r + hl * 8;
            const size_t idx = ((size_t)bh * S_LEN + row) * S_LEN + col;
            const float val = acc[r] * scale + prev[idx]
                            + mask[(size_t)row * S_LEN + col];
            scores[idx] = val;
            acc[r] = val;
            rmax[r] = fmaxf(rmax[r], val);
        }
        sc[nt] = acc;
    }

    // ---------------- stage 2: softmax ----------------------------------
    // reduce max over the 16 cols per lane-group (xor over lane bits 0-3)
#pragma unroll
    for (int off = 8; off >= 1; off >>= 1)
#pragma unroll
        for (int r = 0; r < 8; ++r)
            rmax[r] = fmaxf(rmax[r], __shfl_xor(rmax[r], off, 32));
    if (l16 == 0) {
#pragma unroll
        for (int r = 0; r < 8; ++r)
            redmax[(r + hl * 8) * 8 + wave] = rmax[r];
    }
    __syncthreads();
    float m8[8];
#pragma unroll
    for (int r = 0; r < 8; ++r) {
        const v4f t0 = *(const v4f*)(redmax + (r + hl * 8) * 8);
        const v4f t1 = *(const v4f*)(redmax + (r + hl * 8) * 8 + 4);
        m8[r] = fmaxf(fmaxf(fmaxf(t0.x, t0.y), fmaxf(t0.z, t0.w)),
                      fmaxf(fmaxf(t1.x, t1.y), fmaxf(t1.z, t1.w)));
    }

    float rsum[8];
#pragma unroll
    for (int r = 0; r < 8; ++r) rsum[r] = 0.f;
#pragma unroll
    for (int nt = 0; nt < 8; ++nt)
#pragma unroll
        for (int r = 0; r < 8; ++r) {
            const float e = __expf(sc[nt][r] - m8[r]);
            sc[nt][r] = e;
            rsum[r] += e;
        }
#pragma unroll
    for (int off = 8; off >= 1; off >>= 1)
#pragma unroll
        for (int r = 0; r < 8; ++r)
            rsum[r] += __shfl_xor(rsum[r], off, 32);
    if (l16 == 0) {
#pragma unroll
        for (int r = 0; r < 8; ++r)
            redsum[(r + hl * 8) * 8 + wave] = rsum[r];
    }
    __syncthreads();
    float is8[8];
#pragma unroll
    for (int r = 0; r < 8; ++r) {
        const v4f t0 = *(const v4f*)(redsum + (r + hl * 8) * 8);
        const v4f t1 = *(const v4f*)(redsum + (r + hl * 8) * 8 + 4);
        is8[r] = 1.0f / (((t0.x + t0.y) + (t0.z + t0.w))
                       + ((t1.x + t1.y) + (t1.z + t1.w)));
    }

    // weights: emit to global + stage into LDS (row-major, padded stride)
#pragma unroll
    for (int nt = 0; nt < 8; ++nt) {
        const int col = wave * 128 + nt * 16 + l16;
#pragma unroll
        for (int r = 0; r < 8; ++r) {
            const int row_l = r + hl * 8;
            const float wv = sc[nt][r] * is8[r];
            weights[((size_t)bh * S_LEN + m0 + row_l) * S_LEN + col] = wv;
            lds_w[row_l * LDS_STRIDE + col] = wv;
        }
    }
    __syncthreads();

    // ---------------- stage 3: O = W @ V (split-K over wave pairs) -------
    const float* vbase = v + (size_t)bh * S_LEN * D_HEAD;   // v is [S, D]
    const int ntile = wave & 3;                // output tile (cols of D)
    const int khalf = wave >> 2;               // 0: K 0..511, 1: K 512..1023
    const int colv  = ntile * 16 + l16;
    const float* afr = lds_w + l16 * LDS_STRIDE;            // A = W[l16][*]

    v8f oa0 = {}, oa1 = {}, oa2 = {}, oa3 = {};
    const int kbeg = khalf * 512;
    for (int kb = kbeg; kb < kbeg + 512; kb += 16) {
#pragma unroll
        for (int u = 0; u < 4; ++u) {
            const int kd = kb + u * 4 + hl * 2;
            const v2f a = *(const v2f*)(afr + kd);          // ds_load_b64
            v2f b;
            b.x = vbase[(size_t)kd * D_HEAD + colv];
            b.y = vbase[(size_t)(kd + 1) * D_HEAD + colv];
            v8f& oa = (u == 0) ? oa0 : (u == 1) ? oa1 : (u == 2) ? oa2 : oa3;
            oa = __builtin_amdgcn_wmma_f32_16x16x4_f32(
                false, a, false, b, (short)0, oa, false, false);
        }
    }
    const v8f o = (oa0 + oa1) + (oa2 + oa3);

    // exchange upper-K partials through LDS, combine in fixed order
    if (khalf == 1) {
#pragma unroll
        for (int r = 0; r < 8; ++r)
            obuf[ntile * 256 + (r + hl * 8) * 16 + l16] = o[r];
    }
    __syncthreads();
    if (khalf == 0) {
#pragma unroll
        for (int r = 0; r < 8; ++r) {
            const int row = m0 + r + hl * 8;
            const float tot = o[r] + obuf[ntile * 256 + (r + hl * 8) * 16 + l16];
            out[((size_t)bh * S_LEN + row) * D_HEAD + colv] = tot;
        }
    }
}

// ---------------------------------------------------------------------------
extern "C" void kernel_launch(void* const* d_in, const int* in_sizes, int n_in,
                              void* d_out, int out_size, void* d_ws, size_t ws_size,
                              hipStream_t stream) {
    const float* q     = (const float*)d_in[0];  // [4,16,1024,64]
    const float* k     = (const float*)d_in[1];  // [4,16,64,1024]
    const float* v     = (const float*)d_in[2];  // [4,16,1024,64]
    const float* prev  = (const float*)d_in[3];  // [4,16,1024,1024]
    const float* mask  = (const float*)d_in[4];  // [1,1,1024,1024]
    const float* scale = (const float*)d_in[5];  // scalar

    const size_t BH = 64;                         // 4 * 16
    float* out_o = (float*)d_out;                                   // output
    float* out_w = out_o + BH * (size_t)S_LEN * D_HEAD;             // weights
    float* out_s = out_w + BH * (size_t)S_LEN * S_LEN;              // scores

    // 16*1028 (weight tile) + 2*128 (reductions) + 1024 (split-K partials)
    const size_t shmem = (16 * LDS_STRIDE + 256 + 1024) * sizeof(float);
    dim3 grid(S_LEN / 16, (unsigned)BH);          // (64, 64)
    fused_attn_kernel<<<grid, 256, shmem, stream>>>(
        q, k, v, prev, mask, scale, out_o, out_w, out_s);
}